// MultiHeadSelfAttention_68135361184089
// MI455X (gfx1250) — compile-verified
//
#include <hip/hip_runtime.h>

// Problem constants (B, T, D, H) = (2, 2048, 1024, 16), Dh = 64, fp32.
#define B_  2
#define T_  2048
#define D_  1024
#define H_  16
#define DH_ 64

typedef __attribute__((ext_vector_type(2))) float v2f;
typedef __attribute__((ext_vector_type(8))) float v8f;
typedef unsigned int v4u __attribute__((ext_vector_type(4)));
typedef int v8i __attribute__((ext_vector_type(8)));
typedef int v4i __attribute__((ext_vector_type(4)));

// D = A(16x4,f32) * B(4x16,f32) + C(16x16,f32)  -- wave32, CDNA5 WMMA
__device__ __forceinline__ v8f wmma_f32(v2f a, v2f b, v8f c) {
    return __builtin_amdgcn_wmma_f32_16x16x4_f32(
        /*neg_a=*/false, a, /*neg_b=*/false, b,
        /*c_mod=*/(short)0, c, /*reuse_a=*/false, /*reuse_b=*/false);
}

// ---------------------------------------------------------------------------
// TDM 2D tile load: tile_d0 f32/row x tile_d1 rows, global row stride
// row_stride (f32 units), into LDS with padding: after every
// (2<<pad_int_code) dwords insert (pad_amt_code+1) dwords.
// Wave-scalar async op, tracked by TENSORcnt.
// ---------------------------------------------------------------------------
__device__ __forceinline__ void tdm_load_2d(unsigned int lds_off,
                                            const float* gptr,
                                            unsigned int tile_d0,
                                            unsigned int tile_d1,
                                            unsigned long long row_stride,
                                            unsigned int pad_int_code,
                                            unsigned int pad_amt_code) {
    unsigned long long ga = (unsigned long long)(uintptr_t)gptr;
    v4u g0;
    g0[0] = 1u;                                   // count=1, user descriptor
    g0[1] = lds_off;                              // lds_addr (byte offset)
    g0[2] = (unsigned int)(ga & 0xFFFFFFFFu);     // global_addr[31:0]
    g0[3] = (unsigned int)((ga >> 32) & 0x01FFFFFFu) | (2u << 30);  // addr hi | type=2

    const unsigned int flags = (2u << 16)              // data_size = 4 bytes
                             | (1u << 20)              // pad_enable
                             | (pad_int_code << 22)    // pad interval
                             | (pad_amt_code << 25);   // pad amount
    unsigned long long td0 = tile_d0, td1 = tile_d1;   // tensor dims == tile dims
    unsigned long long q0 = (unsigned long long)flags | ((td0 & 0xFFFFull) << 48);
    unsigned long long q1 = (td0 >> 16) | (td1 << 16) | (td0 << 48);    // tile_dim0
    unsigned long long q2 = td1 | ((row_stride & 0xFFFFFFFFull) << 32); // tile_dim1, stride lo
    unsigned long long q3 = (row_stride >> 32) & 0xFFFFull;             // stride hi
    v8i g1;
    g1[0] = (int)(q0);       g1[1] = (int)(q0 >> 32);
    g1[2] = (int)(q1);       g1[3] = (int)(q1 >> 32);
    g1[4] = (int)(q2);       g1[5] = (int)(q2 >> 32);
    g1[6] = (int)(q3);       g1[7] = (int)(q3 >> 32);
    v4i gz = {};
    v8i gz8 = {};
    __builtin_amdgcn_tensor_load_to_lds(g0, g1, gz, gz, gz8, 0);
}

// ---------------------------------------------------------------------------
// C[M,N] = A[M,K] @ W[N,K]^T + bias[N]
// 256 threads = 8 waves (4 m x 2 n); wave owns 2 m-tiles x 4 n-tiles.
// Block tile 128x128, BK=16, TDM double-buffered staging: wave 0 issues
// tensor_load_to_lds for stage s+2 while everyone computes stage s.
// Per k-step: 6 LDS frag loads -> 8 WMMAs; zero staging vmem in compute waves.
// ---------------------------------------------------------------------------
__global__ __launch_bounds__(256)
void gemm_bias_wmma(const float* __restrict__ A, const float* __restrict__ W,
                    const float* __restrict__ bias, float* __restrict__ C,
                    int M, int N, int K) {
    constexpr int BM = 128, BN = 128, BK = 16, SK = BK + 2; // TDM pad 16+2 dwords
    __shared__ float As[2][BM * SK];
    __shared__ float Ws[2][BN * SK];

    const int tid  = threadIdx.x;
    const int wv   = tid >> 5;
    const int lane = tid & 31;
    const int half = lane >> 4;
    const int ln   = lane & 15;
    const int wm   = wv >> 1;   // 0..3 : m-tiles at rows wm*32 + {0,16}
    const int wn   = wv & 1;    // 0..1 : n-tiles at cols wn*16 + {0,32,64,96}
    const int m0   = blockIdx.y * BM;
    const int n0   = blockIdx.x * BN;
    const int nStages = K / BK;   // 64

    v8f acc[2][4];
    #pragma unroll
    for (int i = 0; i < 2; ++i)
        #pragma unroll
        for (int j = 0; j < 4; ++j) acc[i][j] = (v8f){};

    // prologue: issue TDM for stages 0 and 1 (A and W each)
    if (wv == 0) {
        tdm_load_2d((unsigned int)(uintptr_t)&As[0][0], &A[(size_t)m0 * K],      BK, BM, K, 3, 1);
        tdm_load_2d((unsigned int)(uintptr_t)&Ws[0][0], &W[(size_t)n0 * K],      BK, BN, K, 3, 1);
        tdm_load_2d((unsigned int)(uintptr_t)&As[1][0], &A[(size_t)m0 * K + BK], BK, BM, K, 3, 1);
        tdm_load_2d((unsigned int)(uintptr_t)&Ws[1][0], &W[(size_t)n0 * K + BK], BK, BN, K, 3, 1);
    }

    for (int s = 0; s < nStages; ++s) {
        const int buf = s & 1;
        if (wv == 0) {
            if (s == nStages - 1) __builtin_amdgcn_s_wait_tensorcnt(0);
            else                  __builtin_amdgcn_s_wait_tensorcnt(2);
        }
        __syncthreads();

        const float* as = &As[buf][0];
        const float* ws = &Ws[buf][0];
        #pragma unroll
        for (int kk = 0; kk < BK; kk += 4) {
            const int ks = kk + 2 * half;
            v2f a0 = *(const v2f*)&as[(wm * 32 + ln) * SK + ks];
            v2f a1 = *(const v2f*)&as[(wm * 32 + 16 + ln) * SK + ks];
            v2f b0 = *(const v2f*)&ws[(wn * 16 + ln) * SK + ks];
            v2f b1 = *(const v2f*)&ws[(wn * 16 + 32 + ln) * SK + ks];
            v2f b2 = *(const v2f*)&ws[(wn * 16 + 64 + ln) * SK + ks];
            v2f b3 = *(const v2f*)&ws[(wn * 16 + 96 + ln) * SK + ks];
            acc[0][0] = wmma_f32(a0, b0, acc[0][0]);
            acc[0][1] = wmma_f32(a0, b1, acc[0][1]);
            acc[0][2] = wmma_f32(a0, b2, acc[0][2]);
            acc[0][3] = wmma_f32(a0, b3, acc[0][3]);
            acc[1][0] = wmma_f32(a1, b0, acc[1][0]);
            acc[1][1] = wmma_f32(a1, b1, acc[1][1]);
            acc[1][2] = wmma_f32(a1, b2, acc[1][2]);
            acc[1][3] = wmma_f32(a1, b3, acc[1][3]);
        }
        __syncthreads();

        // issue TDM for stage s+2 into the buffer we just finished reading
        if (wv == 0 && s + 2 < nStages) {
            const int k0 = (s + 2) * BK;
            tdm_load_2d((unsigned int)(uintptr_t)&As[buf][0], &A[(size_t)m0 * K + k0], BK, BM, K, 3, 1);
            tdm_load_2d((unsigned int)(uintptr_t)&Ws[buf][0], &W[(size_t)n0 * K + k0], BK, BN, K, 3, 1);
        }
    }

    // epilogue: C/D layout -> row = r + 8*half, col = ln
    #pragma unroll
    for (int j = 0; j < 4; ++j) {
        const int ng = n0 + wn * 16 + j * 32 + ln;
        const float bb = bias[ng];
        #pragma unroll
        for (int i = 0; i < 2; ++i) {
            #pragma unroll
            for (int r = 0; r < 8; ++r) {
                int mg = m0 + wm * 32 + i * 16 + r + 8 * half;
                C[(size_t)mg * N + ng] = acc[i][j][r] + bb;
            }
        }
    }
}

// ---------------------------------------------------------------------------
// RoPE in-place on q and k thirds of qkv[B,T,3D]. One thread per (b,t,h,pair).
// ---------------------------------------------------------------------------
__global__ __launch_bounds__(256)
void rope_kernel(float* __restrict__ qkv) {
    const int PH  = DH_ / 2;
    int idx = blockIdx.x * blockDim.x + threadIdx.x;
    int i = idx % PH;
    int h = (idx / PH) % H_;
    int t = (idx / (PH * H_)) % T_;
    int b = idx / (PH * H_ * T_);

    float inv = powf(10000.f, (-2.f * (float)i) / (float)DH_);
    float fr  = (float)t * inv;
    float s, c;
    sincosf(fr, &s, &c);

    size_t base = ((size_t)(b * T_ + t)) * (3 * D_) + h * DH_ + 2 * i;
    float qe = qkv[base], qo = qkv[base + 1];
    qkv[base]     = qe * c - qo * s;
    qkv[base + 1] = qe * s + qo * c;
    float ke = qkv[base + D_], ko = qkv[base + D_ + 1];
    qkv[base + D_]     = ke * c - ko * s;
    qkv[base + D_ + 1] = ke * s + ko * c;
}

// ---------------------------------------------------------------------------
// Flash attention (causal), fp32 WMMA.
// grid = (T/64, H, B); block = 128 threads = 4 waves; wave w owns q-tile w.
// Q tile TDM'd once and promoted to registers; its LDS buffer is then reused
// as the P staging area (each wave's P rows == its own Q rows, no hazard).
// K tiles (64 keys) via TDM; V staged transposed so P@V B-frags are float2.
// ---------------------------------------------------------------------------
__global__ __launch_bounds__(128)
void attn_kernel(const float* __restrict__ qkv, float* __restrict__ o) {
    constexpr int QB = 64, KB = 64;
    constexpr int SD  = DH_ + 4;  // Q/K/P row stride (TDM pad 64+4 dwords)
    constexpr int SPV = KB + 6;   // Vt row stride: 8B-aligned, bank-spread
    __shared__ float SQP[QB * SD];       // Q staging, then P tiles
    __shared__ float Ks[KB * SD];
    __shared__ float Vt[DH_ * SPV];      // Vt[dh][key]

    const int tid  = threadIdx.x;
    const int wv   = tid >> 5;
    const int lane = tid & 31;
    const int half = lane >> 4;
    const int ln   = lane & 15;
    const int qb = blockIdx.x, h = blockIdx.y, b = blockIdx.z;
    const int q0 = qb * QB;
    const int qt0 = wv * 16;        // this wave's q rows within block
    const float scale = 0.125f;     // 1/sqrt(Dh=64), applied to raw scores

    // TDM load of the whole Q block (64 rows x 64 cols, stride 3D)
    if (wv == 0) {
        const float* qsrc = &qkv[((size_t)(b * T_ + q0)) * (3 * D_) + h * DH_];
        tdm_load_2d((unsigned int)(uintptr_t)&SQP[0], qsrc, DH_, QB, 3 * D_, 5, 3);
        __builtin_amdgcn_s_wait_tensorcnt(0);
    }
    __syncthreads();

    // promote this wave's Q A-frags to registers (16 x v2f = 32 VGPRs)
    v2f qf[16];
    #pragma unroll
    for (int kq = 0; kq < 16; ++kq)
        qf[kq] = *(const v2f*)&SQP[(qt0 + ln) * SD + kq * 4 + 2 * half];

    float m_i[8], l_i[8];
    #pragma unroll
    for (int r = 0; r < 8; ++r) { m_i[r] = -1e30f; l_i[r] = 0.f; }
    v8f accO[4];
    #pragma unroll
    for (int j = 0; j < 4; ++j) accO[j] = (v8f){};

    const int nKB = qb + 1;         // causal: keys up to q0+63

    for (int kb = 0; kb < nKB; ++kb) {
        const int kr0 = kb * KB;
        __syncthreads();  // previous iteration's readers of Ks/Vt are done

        // K tile via TDM (64 rows x 64 cols, stride 3D -> padded LDS rows)
        if (wv == 0) {
            const float* ksrc = &qkv[((size_t)(b * T_ + kr0)) * (3 * D_) + D_ + h * DH_];
            tdm_load_2d((unsigned int)(uintptr_t)&Ks[0], ksrc, DH_, KB, 3 * D_, 5, 3);
        }
        // V tile staged transposed: Vt[col][row] (64x64 floats, 128 thr x 8 float4)
        #pragma unroll
        for (int it = 0; it < 8; ++it) {
            int flat = (tid + it * 128) * 4;
            int row = flat >> 6, col = flat & 63;
            size_t gb = ((size_t)(b * T_ + kr0 + row)) * (3 * D_) + 2 * D_ + h * DH_ + col;
            float4 v = *(const float4*)&qkv[gb];
            Vt[(col + 0) * SPV + row] = v.x;
            Vt[(col + 1) * SPV + row] = v.y;
            Vt[(col + 2) * SPV + row] = v.z;
            Vt[(col + 3) * SPV + row] = v.w;
        }
        if (wv == 0) __builtin_amdgcn_s_wait_tensorcnt(0);
        __syncthreads();

        // scores S = Q_tile(16xDh) @ K_block^T(Dhx64): four 16x16 WMMA tiles
        v8f S[4];
        #pragma unroll
        for (int st = 0; st < 4; ++st) S[st] = (v8f){};
        #pragma unroll
        for (int kk = 0; kk < DH_; kk += 4) {
            const int ks = kk + 2 * half;
            v2f a = qf[kk >> 2];
            #pragma unroll
            for (int st = 0; st < 4; ++st) {
                v2f bk = *(const v2f*)&Ks[(st * 16 + ln) * SD + ks];
                S[st] = wmma_f32(a, bk, S[st]);
            }
        }

        // scale + causal mask + online softmax (row r+8*half across 16 lanes)
        float alpha[8];
        #pragma unroll
        for (int r = 0; r < 8; ++r) {
            int qIdx = q0 + qt0 + r + 8 * half;
            float rm = -1e30f;
            #pragma unroll
            for (int st = 0; st < 4; ++st) {
                float sv = S[st][r] * scale;
                if (kr0 + st * 16 + ln > qIdx) sv = -1e30f;
                S[st][r] = sv;
                rm = fmaxf(rm, sv);
            }
            rm = fmaxf(rm, __shfl_xor(rm, 8, 32));
            rm = fmaxf(rm, __shfl_xor(rm, 4, 32));
            rm = fmaxf(rm, __shfl_xor(rm, 2, 32));
            rm = fmaxf(rm, __shfl_xor(rm, 1, 32));
            float mn = fmaxf(m_i[r], rm);
            alpha[r] = __expf(m_i[r] - mn);
            m_i[r] = mn;
            float rs = 0.f;
            #pragma unroll
            for (int st = 0; st < 4; ++st) {
                float p = __expf(S[st][r] - mn);
                S[st][r] = p;
                rs += p;
            }
            rs += __shfl_xor(rs, 8, 32);
            rs += __shfl_xor(rs, 4, 32);
            rs += __shfl_xor(rs, 2, 32);
            rs += __shfl_xor(rs, 1, 32);
            l_i[r] = l_i[r] * alpha[r] + rs;
        }
        #pragma unroll
        for (int j = 0; j < 4; ++j)
            #pragma unroll
            for (int r = 0; r < 8; ++r) accO[j][r] *= alpha[r];

        // C-layout -> row-major P into this wave's own rows of SQP
        #pragma unroll
        for (int r = 0; r < 8; ++r) {
            #pragma unroll
            for (int st = 0; st < 4; ++st)
                SQP[(qt0 + r + 8 * half) * SD + st * 16 + ln] = S[st][r];
        }

        // O += P(16x64) @ V(64xDh): B-frags contiguous from transposed Vt
        #pragma unroll
        for (int kk = 0; kk < KB; kk += 4) {
            const int ks = kk + 2 * half;
            v2f a = *(const v2f*)&SQP[(qt0 + ln) * SD + ks];
            #pragma unroll
            for (int j = 0; j < 4; ++j) {
                v2f bv = *(const v2f*)&Vt[(j * 16 + ln) * SPV + ks];
                accO[j] = wmma_f32(a, bv, accO[j]);
            }
        }
    }

    // normalize, write to attn output [B, T, H*Dh]
    #pragma unroll
    for (int r = 0; r < 8; ++r) {
        float invl = 1.f / l_i[r];
        int tq = q0 + qt0 + r + 8 * half;
        size_t base = ((size_t)(b * T_ + tq)) * D_ + h * DH_;
        #pragma unroll
        for (int j = 0; j < 4; ++j)
            o[base + j * 16 + ln] = accO[j][r] * invl;
    }
}

// ---------------------------------------------------------------------------
extern "C" void kernel_launch(void* const* d_in, const int* in_sizes, int n_in,
                              void* d_out, int out_size, void* d_ws, size_t ws_size,
                              hipStream_t stream) {
    const float* x     = (const float*)d_in[0];
    const float* w_qkv = (const float*)d_in[1];
    const float* b_qkv = (const float*)d_in[2];
    const float* w_out = (const float*)d_in[3];
    const float* b_out = (const float*)d_in[4];
    float* out  = (float*)d_out;

    float* qkv  = (float*)d_ws;                         // [B,T,3D]  50.3 MB
    float* attn = qkv + (size_t)B_ * T_ * 3 * D_;       // [B,T,D]   16.8 MB

    const int M = B_ * T_;

    // 1) QKV projection: qkv = x @ w_qkv^T + b_qkv   (M=4096, N=3072, K=1024)
    dim3 g1((3 * D_) / 128, M / 128);
    gemm_bias_wmma<<<g1, 256, 0, stream>>>(x, w_qkv, b_qkv, qkv, M, 3 * D_, D_);

    // 2) RoPE in-place on q,k
    int nrope = B_ * T_ * H_ * (DH_ / 2);
    rope_kernel<<<nrope / 256, 256, 0, stream>>>(qkv);

    // 3) causal flash attention (Q/K via Tensor Data Mover)
    dim3 g3(T_ / 64, H_, B_);
    attn_kernel<<<g3, 128, 0, stream>>>(qkv, attn);

    // 4) output projection: out = attn @ w_out^T + b_out  (M=4096, N=1024, K=1024)
    dim3 g4(D_ / 128, M / 128);
    gemm_bias_wmma<<<g4, 256, 0, stream>>>(attn, w_out, b_out, out, M, D_, D_);
}